// GraphAttentionEmbedding_31241592111233
// MI455X (gfx1250) — compile-verified
//
#include <hip/hip_runtime.h>
#include <hip/hip_bf16.h>
#include <math.h>

typedef __attribute__((ext_vector_type(2))) float v2f;
typedef __attribute__((ext_vector_type(8))) float v8f;

#define NN 50000
#define EE 600000
#define LDSP 130   // padded LDS row stride (floats) -> conflict-free b64 reads

// ---------- workspace layout (bytes) ----------
// Q : NN*128*4 = 25,600,000
// K : 25,600,000
// V : 25,600,000
// e : EE*128*4 = 307,200,000
// alpha/ex : EE*2*4 = 4,800,000
// amax(u32 keys) : NN*2*4 = 400,000
// denom : 400,000
#define WS_Q     0
#define WS_K     (25600000)
#define WS_V     (51200000)
#define WS_E     (76800000)
#define WS_ALPHA (384000000)
#define WS_AMAX  (388800000)
#define WS_DENOM (389200000)

// ordered-float <-> uint monotonic mapping (for atomicMax on f32, incl. negatives)
__device__ __forceinline__ unsigned fkey(float f) {
    unsigned u = __float_as_uint(f);
    return (u & 0x80000000u) ? ~u : (u | 0x80000000u);
}
__device__ __forceinline__ float funkey(unsigned k) {
    unsigned u = (k & 0x80000000u) ? (k & 0x7fffffffu) : ~k;
    return __uint_as_float(u);
}

// One 16x16 output tile: A rows from LDS (16xK=128, stride LDSP), B from global W[128x128].
__device__ __forceinline__ v8f wmma_tile_f32(const float* __restrict__ sA,
                                             const float* __restrict__ W,
                                             int n0, int l, int half) {
    v8f c = {};
    #pragma unroll
    for (int k = 0; k < 128; k += 4) {
        const int kk = k + 2 * half;
        v2f a, b;
        a.x = sA[l * LDSP + kk];
        a.y = sA[l * LDSP + kk + 1];
        b.x = W[(size_t)kk * 128 + n0 + l];
        b.y = W[(size_t)(kk + 1) * 128 + n0 + l];
        c = __builtin_amdgcn_wmma_f32_16x16x4_f32(false, a, false, b,
                                                  (short)0, c, false, false);
    }
    return c;
}

__device__ __forceinline__ void store_tile_bias(v8f c, float* __restrict__ O,
                                                const float* __restrict__ bias,
                                                int m0, int n0, int l, int half) {
    const float bb = bias[n0 + l];
    #pragma unroll
    for (int r = 0; r < 8; ++r) {
        const int m = m0 + r + 8 * half;
        O[(size_t)m * 128 + n0 + l] = c[r] + bb;
    }
}

// ---------------- Kernel 1: node projections (Q,K,V,skip) ----------------
__global__ void node_proj_kernel(const float* __restrict__ x,
                                 const float* __restrict__ Wq, const float* __restrict__ bq,
                                 const float* __restrict__ Wk, const float* __restrict__ bk,
                                 const float* __restrict__ Wv, const float* __restrict__ bv,
                                 const float* __restrict__ Ws, const float* __restrict__ bs,
                                 float* __restrict__ Q, float* __restrict__ K,
                                 float* __restrict__ V, float* __restrict__ outSkip) {
    __shared__ float sA[16 * LDSP];
    const int m0 = blockIdx.x * 16;
    #pragma unroll
    for (int i = 0; i < 8; ++i) {
        const int id  = threadIdx.x + i * 256;
        const int row = id >> 7, col = id & 127;
        sA[row * LDSP + col] = x[(size_t)(m0 + row) * 128 + col];
    }
    __syncthreads();

    const int lane = threadIdx.x & 31;
    const int n0   = (threadIdx.x >> 5) * 16;   // wave -> column tile
    const int half = lane >> 4;
    const int l    = lane & 15;

    v8f c;
    c = wmma_tile_f32(sA, Wq, n0, l, half); store_tile_bias(c, Q,       bq, m0, n0, l, half);
    c = wmma_tile_f32(sA, Wk, n0, l, half); store_tile_bias(c, K,       bk, m0, n0, l, half);
    c = wmma_tile_f32(sA, Wv, n0, l, half); store_tile_bias(c, V,       bv, m0, n0, l, half);
    c = wmma_tile_f32(sA, Ws, n0, l, half); store_tile_bias(c, outSkip, bs, m0, n0, l, half);
}

// ---------------- Kernel 2: edge_attr build + edge GEMM (e = attr @ We) ----------------
__global__ void edge_gemm_kernel(const long long* __restrict__ ei,   // [2,E] int64
                                 const int* __restrict__ last_update,
                                 const int* __restrict__ t,
                                 const float* __restrict__ msg,
                                 const float* __restrict__ tw,
                                 const float* __restrict__ tb,
                                 const float* __restrict__ We,
                                 float* __restrict__ eOut) {
    __shared__ float sA[16 * LDSP];
    const long long e0 = (long long)blockIdx.x * 16;
    #pragma unroll
    for (int i = 0; i < 8; ++i) {
        const int id  = threadIdx.x + i * 256;
        const int row = id >> 7, col = id & 127;
        const long long ed = e0 + row;
        float val;
        if (col < 64) {
            const int s  = (int)ei[ed];                       // src
            const float rt = (float)(last_update[s] - t[ed]); // rel_t
            val = cosf(rt * tw[col] + tb[col]);
        } else {
            val = msg[(size_t)ed * 64 + (col - 64)];
        }
        sA[row * LDSP + col] = val;
    }
    __syncthreads();

    const int lane = threadIdx.x & 31;
    const int n0   = (threadIdx.x >> 5) * 16;
    const int half = lane >> 4;
    const int l    = lane & 15;

    v8f c = wmma_tile_f32(sA, We, n0, l, half);
    #pragma unroll
    for (int r = 0; r < 8; ++r) {
        const long long m = e0 + r + 8 * half;
        eOut[(size_t)m * 128 + n0 + l] = c[r];
    }
}

// ---------------- Kernel 3: init segment buffers ----------------
__global__ void init_seg_kernel(unsigned* __restrict__ amax, float* __restrict__ denom) {
    const int i = blockIdx.x * blockDim.x + threadIdx.x;
    if (i < NN * 2) { amax[i] = 0u; denom[i] = 0.0f; }
}

// ---------------- Kernel 4: alpha + segment max ----------------
__global__ void alpha_kernel(const long long* __restrict__ ei,
                             const float* __restrict__ Q, const float* __restrict__ K,
                             const float* __restrict__ eBuf,
                             float* __restrict__ alpha, unsigned* __restrict__ amax) {
    const int tid = blockIdx.x * blockDim.x + threadIdx.x;
    if (tid >= EE * 2) return;
    const int h = tid & 1, edge = tid >> 1;
    const int src = (int)ei[edge];
    const int dst = (int)ei[EE + edge];
    const float4* qp = (const float4*)(Q    + (size_t)dst  * 128 + h * 64);
    const float4* kp = (const float4*)(K    + (size_t)src  * 128 + h * 64);
    const float4* ep = (const float4*)(eBuf + (size_t)edge * 128 + h * 64);
    float s = 0.0f;
    #pragma unroll
    for (int i = 0; i < 16; ++i) {
        const float4 q = qp[i], k = kp[i], e = ep[i];
        s += q.x * (k.x + e.x) + q.y * (k.y + e.y) + q.z * (k.z + e.z) + q.w * (k.w + e.w);
    }
    const float a = s * 0.125f;     // 1/sqrt(64)
    alpha[tid] = a;
    atomicMax(&amax[dst * 2 + h], fkey(a));
}

// ---------------- Kernel 5: exp + segment sum ----------------
__global__ void exden_kernel(const long long* __restrict__ ei,
                             float* __restrict__ alpha,   // in: alpha, out: ex
                             const unsigned* __restrict__ amax,
                             float* __restrict__ denom) {
    const int tid = blockIdx.x * blockDim.x + threadIdx.x;
    if (tid >= EE * 2) return;
    const int h = tid & 1, edge = tid >> 1;
    const int dst = (int)ei[EE + edge];
    const float m  = funkey(amax[dst * 2 + h]);
    const float ex = expf(alpha[tid] - m);
    alpha[tid] = ex;
    unsafeAtomicAdd(&denom[dst * 2 + h], ex);
}

// ---------------- Kernel 6: weighted scatter-add ----------------
__global__ void scatter_kernel(const long long* __restrict__ ei,
                               const float* __restrict__ V,
                               const float* __restrict__ eBuf,
                               const float* __restrict__ ex,
                               const float* __restrict__ denom,
                               float* __restrict__ out) {
    const int tid = blockIdx.x * blockDim.x + threadIdx.x;
    if (tid >= EE * 2) return;
    const int h = tid & 1, edge = tid >> 1;
    const int src = (int)ei[edge];
    const int dst = (int)ei[EE + edge];
    const float attn = ex[tid] / (denom[dst * 2 + h] + 1e-16f);
    const float4* vp = (const float4*)(V    + (size_t)src  * 128 + h * 64);
    const float4* ep = (const float4*)(eBuf + (size_t)edge * 128 + h * 64);
    float* op = out + (size_t)dst * 128 + h * 64;
    #pragma unroll
    for (int i = 0; i < 16; ++i) {
        const float4 v = vp[i], e = ep[i];
        unsafeAtomicAdd(op + 4 * i + 0, (v.x + e.x) * attn);
        unsafeAtomicAdd(op + 4 * i + 1, (v.y + e.y) * attn);
        unsafeAtomicAdd(op + 4 * i + 2, (v.z + e.z) * attn);
        unsafeAtomicAdd(op + 4 * i + 3, (v.w + e.w) * attn);
    }
}

extern "C" void kernel_launch(void* const* d_in, const int* in_sizes, int n_in,
                              void* d_out, int out_size, void* d_ws, size_t ws_size,
                              hipStream_t stream) {
    const float*     x           = (const float*)d_in[0];
    const int*       last_update = (const int*)d_in[1];
    const long long* ei          = (const long long*)d_in[2];
    const int*       t           = (const int*)d_in[3];
    const float*     msg         = (const float*)d_in[4];
    const float*     tw          = (const float*)d_in[5];
    const float*     tb          = (const float*)d_in[6];
    const float*     Wq = (const float*)d_in[7];
    const float*     bq = (const float*)d_in[8];
    const float*     Wk = (const float*)d_in[9];
    const float*     bk = (const float*)d_in[10];
    const float*     Wv = (const float*)d_in[11];
    const float*     bv = (const float*)d_in[12];
    const float*     We = (const float*)d_in[13];
    const float*     Ws = (const float*)d_in[14];
    const float*     bs = (const float*)d_in[15];
    float* out = (float*)d_out;

    char* ws = (char*)d_ws;
    float*    Qb    = (float*)(ws + WS_Q);
    float*    Kb    = (float*)(ws + WS_K);
    float*    Vb    = (float*)(ws + WS_V);
    float*    Eb    = (float*)(ws + WS_E);
    float*    Ab    = (float*)(ws + WS_ALPHA);
    unsigned* AMb   = (unsigned*)(ws + WS_AMAX);
    float*    Db    = (float*)(ws + WS_DENOM);

    init_seg_kernel<<<(NN * 2 + 255) / 256, 256, 0, stream>>>(AMb, Db);
    node_proj_kernel<<<NN / 16, 256, 0, stream>>>(x, Wq, bq, Wk, bk, Wv, bv, Ws, bs,
                                                  Qb, Kb, Vb, out);
    edge_gemm_kernel<<<EE / 16, 256, 0, stream>>>(ei, last_update, t, msg, tw, tb, We, Eb);
    alpha_kernel<<<(EE * 2 + 255) / 256, 256, 0, stream>>>(ei, Qb, Kb, Eb, Ab, AMb);
    exden_kernel<<<(EE * 2 + 255) / 256, 256, 0, stream>>>(ei, Ab, AMb, Db);
    scatter_kernel<<<(EE * 2 + 255) / 256, 256, 0, stream>>>(ei, Vb, Eb, Ab, Db, out);
}